// SGNN_27058293965169
// MI455X (gfx1250) — compile-verified
//
#include <hip/hip_runtime.h>
#include <math.h>

typedef float v8f __attribute__((ext_vector_type(8)));
typedef float v2f __attribute__((ext_vector_type(2)));

// ---------- helpers ----------
__device__ __forceinline__ unsigned f2ord(float f) {
  unsigned u = __float_as_uint(f);
  return (u & 0x80000000u) ? ~u : (u | 0x80000000u);
}
__device__ __forceinline__ float ord2f(unsigned u) {
  return (u & 0x80000000u) ? __uint_as_float(u & 0x7fffffffu)
                           : __uint_as_float(~u);
}

// ---------- fills ----------
__global__ void k_fill_f(float* p, float v, int n) {
  int i = blockIdx.x * 256 + threadIdx.x;
  if (i < n) p[i] = v;
}
__global__ void k_fill_u(unsigned* p, unsigned v, int n) {
  int i = blockIdx.x * 256 + threadIdx.x;
  if (i < n) p[i] = v;
}

// ---------- generic fp32 WMMA GEMM: C[M,Nout] = A[M,K] @ W[K,Nout] (+bias) ----
// grid: (M/16, ceil(Nout/64)); block 128 (4 waves); wave w -> 16-col tile w.
// A-strip (16 x K) staged in LDS (padded stride K+1). K must be multiple of 4.
__global__ __launch_bounds__(128)
void k_gemm_wmma(const float* __restrict__ A, const float* __restrict__ W,
                 const float* __restrict__ bias, float* __restrict__ C,
                 int M, int K, int Nout) {
  __shared__ float Alds[16 * 257];
  const int ld = K + 1;
  const int tid = threadIdx.x;
  const int rowBase = blockIdx.x * 16;
  for (int idx = tid; idx < 16 * K; idx += 128) {
    int r = idx / K, k = idx - r * K;
    int row = rowBase + r;
    Alds[r * ld + k] = (row < M) ? A[(size_t)row * K + k] : 0.f;
  }
  __syncthreads();

  const int wave = tid >> 5, lane = tid & 31;
  const int m = lane & 15;                 // A row within tile
  const int koff = (lane >> 4) << 1;       // lanes 0-15: K0,K1  lanes 16-31: K2,K3
  const int ncol = blockIdx.y * 64 + wave * 16 + (lane & 15);
  const bool cok = ncol < Nout;
  const int nc = cok ? ncol : 0;

  v8f acc = {};
  for (int k = 0; k < K; k += 4) {
    v2f a, b;
    a.x = Alds[m * ld + k + koff];
    a.y = Alds[m * ld + k + koff + 1];
    float b0 = W[(size_t)(k + koff) * Nout + nc];
    float b1 = W[(size_t)(k + koff + 1) * Nout + nc];
    b.x = cok ? b0 : 0.f;
    b.y = cok ? b1 : 0.f;
    acc = __builtin_amdgcn_wmma_f32_16x16x4_f32(false, a, false, b,
                                                (short)0, acc, false, false);
  }
  if (cok) {
    float bv = bias ? bias[ncol] : 0.f;
    int rowTop = rowBase + ((lane >> 4) << 3);   // lanes>=16 hold M=8..15
#pragma unroll
    for (int r = 0; r < 8; ++r) {
      int row = rowTop + r;
      if (row < M) C[(size_t)row * Nout + ncol] = acc[r] + bv;
    }
  }
}

// ---------- GAT attention coefficients: s[n,h], d[n,h] (C=64) ----------
__global__ void k_gat_sd(const float* __restrict__ h,
                         const float* __restrict__ a_src,
                         const float* __restrict__ a_dst,
                         float* __restrict__ s, float* __restrict__ d,
                         int N, int Hn) {
  int gid = blockIdx.x * 256 + threadIdx.x;
  if (gid >= N * Hn) return;
  int n = gid / Hn, hh = gid - n * Hn;
  const float* hp = h + (size_t)n * Hn * 64 + hh * 64;
  const float* as = a_src + hh * 64;
  const float* ad = a_dst + hh * 64;
  float ss = 0.f, dd = 0.f;
#pragma unroll 4
  for (int c = 0; c < 64; ++c) { ss += hp[c] * as[c]; dd += hp[c] * ad[c]; }
  s[gid] = ss;
  d[gid] = dd;
}

// ---------- edge pass 1: segment-max of leaky_relu logits ----------
__global__ void k_gat_max(const int* __restrict__ src, const int* __restrict__ dst,
                          int E, int ET, int Hn,
                          const float* __restrict__ s, const float* __restrict__ d,
                          unsigned* __restrict__ menc) {
  int gid = blockIdx.x * 256 + threadIdx.x;
  if (gid >= ET * Hn) return;
  int e = gid / Hn, h = gid - e * Hn;
  int u = (e < E) ? src[e] : (e - E);
  int v = (e < E) ? dst[e] : (e - E);
  float ev = s[u * Hn + h] + d[v * Hn + h];
  ev = ev > 0.f ? ev : 0.2f * ev;
  atomicMax(&menc[v * Hn + h], f2ord(ev));
}

// ---------- edge pass 2: ex = exp(e - max), denom += ex ----------
__global__ void k_gat_ex(const int* __restrict__ src, const int* __restrict__ dst,
                         int E, int ET, int Hn,
                         const float* __restrict__ s, const float* __restrict__ d,
                         const unsigned* __restrict__ menc,
                         float* __restrict__ exb, float* __restrict__ dn) {
  int gid = blockIdx.x * 256 + threadIdx.x;
  if (gid >= ET * Hn) return;
  int e = gid / Hn, h = gid - e * Hn;
  int u = (e < E) ? src[e] : (e - E);
  int v = (e < E) ? dst[e] : (e - E);
  float ev = s[u * Hn + h] + d[v * Hn + h];
  ev = ev > 0.f ? ev : 0.2f * ev;
  float ex = __expf(ev - ord2f(menc[v * Hn + h]));
  exb[gid] = ex;
  atomicAdd(&dn[v * Hn + h], ex);
}

// ---------- edge pass 3: acc[dst] += ex * feat[src]  (float4 channels) -------
__global__ void k_gat_scatter(const int* __restrict__ src, const int* __restrict__ dst,
                              int E, int ET, int Hn,
                              const float* __restrict__ exb,
                              const float* __restrict__ feat,
                              float* __restrict__ acc) {
  int G = Hn * 16;                                   // float4 groups per row
  int gid = blockIdx.x * 256 + threadIdx.x;
  if (gid >= ET * G) return;
  int e = gid / G, f4 = gid - e * G;
  int u = (e < E) ? src[e] : (e - E);
  int v = (e < E) ? dst[e] : (e - E);
  int h = f4 >> 4;
  float w = exb[e * Hn + h];
  float4 fv = ((const float4*)feat)[(size_t)u * G + f4];
  float* o = acc + (size_t)v * (G * 4) + f4 * 4;
  atomicAdd(o + 0, w * fv.x);
  atomicAdd(o + 1, w * fv.y);
  atomicAdd(o + 2, w * fv.z);
  atomicAdd(o + 3, w * fv.w);
}

// ---------- normalize by softmax denominator ----------
__global__ void k_gat_fin(float* __restrict__ acc, const float* __restrict__ dn,
                          int N, int Hn) {
  int F = Hn * 64;
  int gid = blockIdx.x * 256 + threadIdx.x;
  if (gid >= N * F) return;
  int n = gid / F, f = gid - n * F;
  acc[gid] = acc[gid] / dn[n * Hn + (f >> 6)];
}

// ---------- GCN degree / dinv / scatter ----------
__global__ void k_deg(const int* __restrict__ dst, int E, int ET,
                      float* __restrict__ deg) {
  int gid = blockIdx.x * 256 + threadIdx.x;
  if (gid >= ET) return;
  int v = (gid < E) ? dst[gid] : (gid - E);
  atomicAdd(&deg[v], 1.f);
}
__global__ void k_rsqrt(const float* __restrict__ deg, float* __restrict__ dinv,
                        int N) {
  int i = blockIdx.x * 256 + threadIdx.x;
  if (i < N) dinv[i] = rsqrtf(deg[i]);
}
__global__ void k_gcn_scatter(const int* __restrict__ src, const int* __restrict__ dst,
                              int E, int ET, const float* __restrict__ dinv,
                              const float* __restrict__ feat, float* __restrict__ acc) {
  int gid = blockIdx.x * 256 + threadIdx.x;
  if (gid >= ET * 16) return;
  int e = gid >> 4, f4 = gid & 15;
  int u = (e < E) ? src[e] : (e - E);
  int v = (e < E) ? dst[e] : (e - E);
  float w = dinv[u] * dinv[v];
  float4 fv = ((const float4*)feat)[(size_t)u * 16 + f4];
  float* o = acc + (size_t)v * 64 + f4 * 4;
  atomicAdd(o + 0, w * fv.x);
  atomicAdd(o + 1, w * fv.y);
  atomicAdd(o + 2, w * fv.z);
  atomicAdd(o + 3, w * fv.w);
}

// ---------- BatchNorm: column sums / apply+ELU ----------
__global__ __launch_bounds__(256)
void k_bn_stats(const float* __restrict__ x, float* __restrict__ sums,
                int N, int C) {
  int c = threadIdx.x % C;
  int rl = threadIdx.x / C;
  int rep = 256 / C;
  int r0 = blockIdx.x * 128;
  int r1 = r0 + 128; if (r1 > N) r1 = N;
  float s = 0.f, q = 0.f;
  for (int r = r0 + rl; r < r1; r += rep) {
    float v = x[(size_t)r * C + c];
    s += v; q += v * v;
  }
  atomicAdd(&sums[c], s);
  atomicAdd(&sums[C + c], q);
}
__global__ void k_bn_apply(float* __restrict__ x, const float* __restrict__ sums,
                           const float* __restrict__ g, const float* __restrict__ b,
                           int total, int C, float invN) {
  int gid = blockIdx.x * 256 + threadIdx.x;
  if (gid >= total) return;
  int c = gid % C;
  float mean = sums[c] * invN;
  float var = sums[C + c] * invN - mean * mean;
  float y = (x[gid] - mean) * rsqrtf(var + 1e-5f) * g[c] + b[c];
  x[gid] = y > 0.f ? y : expm1f(y);
}

// ---------- concat [t | i] -> c (64 + 64 = 128) ----------
__global__ void k_concat(const float* __restrict__ a, const float* __restrict__ b,
                         float* __restrict__ c, int N) {
  int gid = blockIdx.x * 256 + threadIdx.x;
  if (gid >= N * 128) return;
  int n = gid >> 7, f = gid & 127;
  c[gid] = (f < 64) ? a[(size_t)n * 64 + f] : b[(size_t)n * 64 + f - 64];
}

// ---------- per-node MHA: 4 heads x 16 dims, warp per node ----------
__global__ __launch_bounds__(256)
void k_mha(const float* __restrict__ q, const float* __restrict__ k,
           const float* __restrict__ v, float* __restrict__ out, int N) {
  int warp = threadIdx.x >> 5, lane = threadIdx.x & 31;
  int n = blockIdx.x * 8 + warp;
  if (n >= N) return;
  const float* qr = q + (size_t)n * 64;
  const float* kr = k + (size_t)n * 64;
  const float* vr = v + (size_t)n * 64;
  int h = (lane >> 2) & 3, g = lane & 3;          // lanes 16-31 mirror 0-15
  float sc = 0.f;
#pragma unroll
  for (int dd = 0; dd < 16; ++dd) sc += qr[h * 16 + dd] * kr[g * 16 + dd];
  sc *= 0.25f;                                    // 1/sqrt(16)
  float mx = fmaxf(sc, __shfl_xor(sc, 1));
  mx = fmaxf(mx, __shfl_xor(mx, 2));
  float e = __expf(sc - mx);
  float ssum = e + __shfl_xor(e, 1);
  ssum += __shfl_xor(ssum, 2);
  float attn = e / ssum;
#pragma unroll
  for (int t = 0; t < 2; ++t) {
    int idx = lane + t * 32;
    int oh = idx >> 4, od = idx & 15;
    float o = 0.f;
#pragma unroll
    for (int gg = 0; gg < 4; ++gg)
      o += __shfl(attn, oh * 4 + gg) * vr[gg * 16 + od];
    out[(size_t)n * 64 + idx] = o;
  }
}

// ---------- temperature (x2) log-softmax over 40 classes, warp per node -----
__global__ __launch_bounds__(256)
void k_lsm(const float* __restrict__ lg, float* __restrict__ out, int N) {
  int warp = threadIdx.x >> 5, lane = threadIdx.x & 31;
  int n = blockIdx.x * 8 + warp;
  if (n >= N) return;
  const float* r = lg + (size_t)n * 40;
  float v0 = 2.f * r[lane];
  float v1 = (lane < 8) ? 2.f * r[32 + lane] : -INFINITY;
  float m = fmaxf(v0, v1);
  for (int o = 16; o; o >>= 1) m = fmaxf(m, __shfl_xor(m, o));
  float s = __expf(v0 - m) + ((lane < 8) ? __expf(v1 - m) : 0.f);
  for (int o = 16; o; o >>= 1) s += __shfl_xor(s, o);
  float ls = m + logf(s);
  out[(size_t)n * 40 + lane] = v0 - ls;
  if (lane < 8) out[(size_t)n * 40 + 32 + lane] = v1 - ls;
}

// ---------- host ----------
static inline int cdiv(long long a, long long b) { return (int)((a + b - 1) / b); }

static void gemm(const float* A, const float* W, const float* bias, float* C,
                 int M, int K, int Nout, hipStream_t s) {
  dim3 g(cdiv(M, 16), cdiv(Nout, 64));
  k_gemm_wmma<<<g, 128, 0, s>>>(A, W, bias, C, M, K, Nout);
}
static void fillf(float* p, float v, long long n, hipStream_t s) {
  k_fill_f<<<cdiv(n, 256), 256, 0, s>>>(p, v, (int)n);
}
static void fillu(unsigned* p, unsigned v, long long n, hipStream_t s) {
  k_fill_u<<<cdiv(n, 256), 256, 0, s>>>(p, v, (int)n);
}

extern "C" void kernel_launch(void* const* d_in, const int* in_sizes, int n_in,
                              void* d_out, int out_size, void* d_ws, size_t ws_size,
                              hipStream_t stream) {
  (void)n_in; (void)out_size; (void)ws_size;
  const float* x      = (const float*)d_in[0];
  const int*   ei     = (const int*)d_in[1];
  const float* W_gat1 = (const float*)d_in[2];
  const float* a_src1 = (const float*)d_in[3];
  const float* a_dst1 = (const float*)d_in[4];
  const float* W_gat2 = (const float*)d_in[6];
  const float* a_src2 = (const float*)d_in[7];
  const float* a_dst2 = (const float*)d_in[8];
  const float* W_gcn1 = (const float*)d_in[10];
  const float* W_gcn2 = (const float*)d_in[12];
  const float* g_tbn1 = (const float*)d_in[14]; const float* be_tbn1 = (const float*)d_in[15];
  const float* g_tbn2 = (const float*)d_in[16]; const float* be_tbn2 = (const float*)d_in[17];
  const float* g_ibn1 = (const float*)d_in[18]; const float* be_ibn1 = (const float*)d_in[19];
  const float* g_ibn2 = (const float*)d_in[20]; const float* be_ibn2 = (const float*)d_in[21];
  const float* W_fus  = (const float*)d_in[22];
  const float* g_fbn  = (const float*)d_in[24]; const float* be_fbn  = (const float*)d_in[25];
  const float* Wq = (const float*)d_in[26]; const float* bq = (const float*)d_in[27];
  const float* Wk = (const float*)d_in[28]; const float* bk = (const float*)d_in[29];
  const float* Wv = (const float*)d_in[30]; const float* bv = (const float*)d_in[31];
  const float* Wp = (const float*)d_in[32];
  const float* Wc1 = (const float*)d_in[34];
  const float* g_cbn = (const float*)d_in[36]; const float* be_cbn = (const float*)d_in[37];
  const float* Wc2 = (const float*)d_in[38]; const float* bc2 = (const float*)d_in[39];
  float* out = (float*)d_out;

  const int N = in_sizes[0] / 128;
  const int E = in_sizes[1] / 2;
  const int ET = E + N;
  const int* esrc = ei;
  const int* edst = ei + E;
  const float invN = 1.f / (float)N;
  const unsigned NEG_INF_ORD = 0x007FFFFFu;   // f2ord(-inf)

  // ----- workspace layout (floats) -----
  float* wsf = (float*)d_ws;
  size_t off = 0;
  float* h1   = wsf + off; off += (size_t)N * 256;  // reused: h2 | t2 | cbuf
  float* tbuf = wsf + off; off += (size_t)N * 256;  // reused: a2 | hb | lg
  float* gbuf = wsf + off; off += (size_t)N * 256;  // 4 x N*64 slots
  float* s1  = wsf + off; off += (size_t)N * 4;
  float* d1  = wsf + off; off += (size_t)N * 4;
  float* m1  = wsf + off; off += (size_t)N * 4;     // used as unsigned
  float* dn1 = wsf + off; off += (size_t)N * 4;
  float* s2  = wsf + off; off += N;
  float* d2  = wsf + off; off += N;
  float* m2  = wsf + off; off += N;                 // used as unsigned
  float* dn2 = wsf + off; off += N;
  float* deg  = wsf + off; off += N;
  float* dinv = wsf + off; off += N;
  float* exb  = wsf + off; off += (size_t)ET * 4;
  float* sums = wsf + off; off += 512;

  float* h2  = h1;                     // [N,64]
  float* t2  = h1 + (size_t)N * 64;    // [N,64]
  float* cbf = h1 + (size_t)N * 128;   // [N,128]
  float* a2  = tbuf;                   // [N,64]
  float* hb  = tbuf + (size_t)N * 64;  // [N,64]
  float* lg  = tbuf + (size_t)N * 128; // [N,40]
  float* gb0 = gbuf;                   // g1 -> f -> mo
  float* gb1 = gbuf + (size_t)N * 64;  // i1 -> q
  float* gb2 = gbuf + (size_t)N * 128; // g2 -> k
  float* gb3 = gbuf + (size_t)N * 192; // i2 -> v

  // ================= trans branch: GAT1 (4 heads) =================
  gemm(x, W_gat1, nullptr, h1, N, 128, 256, stream);
  k_gat_sd<<<cdiv((long long)N * 4, 256), 256, 0, stream>>>(h1, a_src1, a_dst1, s1, d1, N, 4);
  fillu((unsigned*)m1, NEG_INF_ORD, (long long)N * 4, stream);
  fillf(dn1, 0.f, (long long)N * 4, stream);
  fillf(tbuf, 0.f, (long long)N * 256, stream);
  k_gat_max<<<cdiv((long long)ET * 4, 256), 256, 0, stream>>>(esrc, edst, E, ET, 4, s1, d1, (unsigned*)m1);
  k_gat_ex<<<cdiv((long long)ET * 4, 256), 256, 0, stream>>>(esrc, edst, E, ET, 4, s1, d1, (unsigned*)m1, exb, dn1);
  k_gat_scatter<<<cdiv((long long)ET * 64, 256), 256, 0, stream>>>(esrc, edst, E, ET, 4, exb, h1, tbuf);
  k_gat_fin<<<cdiv((long long)N * 256, 256), 256, 0, stream>>>(tbuf, dn1, N, 4);
  fillf(sums, 0.f, 512, stream);
  k_bn_stats<<<cdiv(N, 128), 256, 0, stream>>>(tbuf, sums, N, 256);
  k_bn_apply<<<cdiv((long long)N * 256, 256), 256, 0, stream>>>(tbuf, sums, g_tbn1, be_tbn1, N * 256, 256, invN);

  // ================= GAT2 (1 head, mean==identity since heads=1) ==========
  gemm(tbuf, W_gat2, nullptr, h2, N, 256, 64, stream);
  k_gat_sd<<<cdiv(N, 256), 256, 0, stream>>>(h2, a_src2, a_dst2, s2, d2, N, 1);
  fillu((unsigned*)m2, NEG_INF_ORD, N, stream);
  fillf(dn2, 0.f, N, stream);
  fillf(t2, 0.f, (long long)N * 64, stream);
  k_gat_max<<<cdiv(ET, 256), 256, 0, stream>>>(esrc, edst, E, ET, 1, s2, d2, (unsigned*)m2);
  k_gat_ex<<<cdiv(ET, 256), 256, 0, stream>>>(esrc, edst, E, ET, 1, s2, d2, (unsigned*)m2, exb, dn2);
  k_gat_scatter<<<cdiv((long long)ET * 16, 256), 256, 0, stream>>>(esrc, edst, E, ET, 1, exb, h2, t2);
  k_gat_fin<<<cdiv((long long)N * 64, 256), 256, 0, stream>>>(t2, dn2, N, 1);
  fillf(sums, 0.f, 512, stream);
  k_bn_stats<<<cdiv(N, 128), 256, 0, stream>>>(t2, sums, N, 64);
  k_bn_apply<<<cdiv((long long)N * 64, 256), 256, 0, stream>>>(t2, sums, g_tbn2, be_tbn2, N * 64, 64, invN);

  // ================= ident branch: GCN1 / GCN2 =================
  fillf(deg, 0.f, N, stream);
  k_deg<<<cdiv(ET, 256), 256, 0, stream>>>(edst, E, ET, deg);
  k_rsqrt<<<cdiv(N, 256), 256, 0, stream>>>(deg, dinv, N);

  gemm(x, W_gcn1, nullptr, gb0, N, 128, 64, stream);
  fillf(gb1, 0.f, (long long)N * 64, stream);
  k_gcn_scatter<<<cdiv((long long)ET * 16, 256), 256, 0, stream>>>(esrc, edst, E, ET, dinv, gb0, gb1);
  fillf(sums, 0.f, 512, stream);
  k_bn_stats<<<cdiv(N, 128), 256, 0, stream>>>(gb1, sums, N, 64);
  k_bn_apply<<<cdiv((long long)N * 64, 256), 256, 0, stream>>>(gb1, sums, g_ibn1, be_ibn1, N * 64, 64, invN);

  gemm(gb1, W_gcn2, nullptr, gb2, N, 64, 64, stream);
  fillf(gb3, 0.f, (long long)N * 64, stream);
  k_gcn_scatter<<<cdiv((long long)ET * 16, 256), 256, 0, stream>>>(esrc, edst, E, ET, dinv, gb2, gb3);
  fillf(sums, 0.f, 512, stream);
  k_bn_stats<<<cdiv(N, 128), 256, 0, stream>>>(gb3, sums, N, 64);
  k_bn_apply<<<cdiv((long long)N * 64, 256), 256, 0, stream>>>(gb3, sums, g_ibn2, be_ibn2, N * 64, 64, invN);

  // ================= fusion -> MHA -> classifier =================
  k_concat<<<cdiv((long long)N * 128, 256), 256, 0, stream>>>(t2, gb3, cbf, N);
  gemm(cbf, W_fus, nullptr, gb0, N, 128, 64, stream);      // f
  fillf(sums, 0.f, 512, stream);
  k_bn_stats<<<cdiv(N, 128), 256, 0, stream>>>(gb0, sums, N, 64);
  k_bn_apply<<<cdiv((long long)N * 64, 256), 256, 0, stream>>>(gb0, sums, g_fbn, be_fbn, N * 64, 64, invN);

  gemm(gb0, Wq, bq, gb1, N, 64, 64, stream);               // q
  gemm(gb0, Wk, bk, gb2, N, 64, 64, stream);               // k
  gemm(gb0, Wv, bv, gb3, N, 64, 64, stream);               // v
  k_mha<<<cdiv(N, 8), 256, 0, stream>>>(gb1, gb2, gb3, gb0, N);   // mo -> gb0
  gemm(gb0, Wp, nullptr, a2, N, 64, 64, stream);           // a (bp cancels in BN)

  gemm(a2, Wc1, nullptr, hb, N, 64, 64, stream);           // (bc1 cancels in BN)
  fillf(sums, 0.f, 512, stream);
  k_bn_stats<<<cdiv(N, 128), 256, 0, stream>>>(hb, sums, N, 64);
  k_bn_apply<<<cdiv((long long)N * 64, 256), 256, 0, stream>>>(hb, sums, g_cbn, be_cbn, N * 64, 64, invN);

  gemm(hb, Wc2, bc2, lg, N, 64, 40, stream);               // pre-temperature logits
  k_lsm<<<cdiv(N, 8), 256, 0, stream>>>(lg, out, N);
}